// ColorsRasterizer_1108101563061
// MI455X (gfx1250) — compile-verified
//
#include <hip/hip_runtime.h>
#include <hip/hip_bf16.h>

typedef __attribute__((ext_vector_type(2))) float v2f;
typedef __attribute__((ext_vector_type(8))) float v8f;

#define RH 256
#define RW 256
#define RC 8

#define GROUPS_PER_CHUNK 16                       // 16 groups x 16 tris = 256 tris/chunk
#define FUNC_BYTES       4096                     // 16 groups * 256B per function
#define NFUNC            4                        // w0, w1, w2, depth
#define CHUNK_BYTES      (NFUNC * FUNC_BYTES)     // 16 KB
#define BUF_BYTES        (2 * CHUNK_BYTES)        // 32 KB double buffer

// ---------------------------------------------------------------------------
// Pass 1: per-triangle setup. Four affine records (A,B,C,0):
//   w0 = A0*x+B0*y+C0, w1 = ..., w2 = 1-w0-w1 (own record), depth = Az*x+Bz*y+Cz
// Degenerate / behind-camera triangles get w0 == -1 everywhere and depth == 0,
// so they can never pass (min3(w)>=0 && depth>0).
// ---------------------------------------------------------------------------
__global__ void raster_pre(const float* __restrict__ verts,
                           const int*   __restrict__ faces,
                           float4* __restrict__ e0,
                           float4* __restrict__ e1,
                           float4* __restrict__ e2,
                           float4* __restrict__ ez,
                           int N, int V, int F, int Fpad)
{
    int t = blockIdx.x * blockDim.x + threadIdx.x;
    if (t >= N * Fpad) return;
    int n = t / Fpad;
    int f = t - n * Fpad;

    float4 r0 = make_float4(0.f, 0.f, -1.f, 0.f);
    float4 r1 = make_float4(0.f, 0.f, -1.f, 0.f);
    float4 r2 = make_float4(0.f, 0.f,  0.f, 0.f);
    float4 rz = make_float4(0.f, 0.f,  0.f, 0.f);

    if (f < F) {
        int i0 = faces[f * 3 + 0];
        int i1 = faces[f * 3 + 1];
        int i2 = faces[f * 3 + 2];
        const float* p0 = verts + ((size_t)n * V + i0) * 3;
        const float* p1 = verts + ((size_t)n * V + i1) * 3;
        const float* p2 = verts + ((size_t)n * V + i2) * 3;
        float z0 = p0[2], z1 = p1[2], z2 = p2[2];
        float x0 = p0[0] / z0, y0 = p0[1] / z0;
        float x1 = p1[0] / z1, y1 = p1[1] / z1;
        float x2 = p2[0] / z2, y2 = p2[1] / z2;
        float denom = (y1 - y2) * (x0 - x2) + (x2 - x1) * (y0 - y2);
        bool valid = (fabsf(denom) > 1e-8f) && (z0 > 0.f) && (z1 > 0.f) && (z2 > 0.f);
        if (valid) {
            float inv = 1.0f / denom;
            float A0 = (y1 - y2) * inv, B0 = (x2 - x1) * inv;
            float C0 = -(A0 * x2 + B0 * y2);
            float A1 = (y2 - y0) * inv, B1 = (x0 - x2) * inv;
            float C1 = -(A1 * x2 + B1 * y2);
            float dz0 = z0 - z2, dz1 = z1 - z2;
            r0 = make_float4(A0, B0, C0, 0.f);
            r1 = make_float4(A1, B1, C1, 0.f);
            r2 = make_float4(-(A0 + A1), -(B0 + B1), 1.0f - C0 - C1, 0.f);
            rz = make_float4(A0 * dz0 + A1 * dz1,
                             B0 * dz0 + B1 * dz1,
                             C0 * dz0 + C1 * dz1 + z2, 0.f);
        }
    }
    e0[t] = r0;
    e1[t] = r1;
    e2[t] = r2;
    ez[t] = rz;
}

// ---------------------------------------------------------------------------
// Pass 2: WMMA rasterizer with double-buffered async LDS staging.
// Each block (8 waves) shares one coefficient stream: per 256-triangle chunk,
// 16 KB of (w0,w1,w2,depth) records is DMA'd into LDS with
// global_load_async_to_lds_b128 while the previous chunk is rasterized.
// Each wave owns 16 pixels of one row; 4 WMMAs evaluate 4 affine functions for
// 16 pixels x 16 triangles; per-lane z-test, then shfl_xor butterfly reduces
// (depth, id) across the 16 triangle columns.
// ---------------------------------------------------------------------------
__global__ void __launch_bounds__(256)
raster_main(const float* __restrict__ colors,
            const int*   __restrict__ faces,
            const float4* __restrict__ e0,
            const float4* __restrict__ e1,
            const float4* __restrict__ e2,
            const float4* __restrict__ ez,
            float* __restrict__ out,
            int V, int Fpad)
{
    __shared__ __align__(16) char smem[BUF_BYTES];
    __shared__ float sd[8][16];
    __shared__ int   si[8][16];

    const int t    = threadIdx.x;
    const int lane = t & 31;
    const int wave = t >> 5;                 // 8 waves per block
    const int col  = lane & 15;              // triangle column
    const int hi   = lane >> 4;              // lane half selects K-pair
    const int n    = blockIdx.z;
    const int row  = blockIdx.y * 8 + wave;
    const int x0   = blockIdx.x * 16;

    const int chunks = Fpad >> 8;            // 256 triangles per chunk

    // Per-function global byte bases for image n.
    const char* g0 = (const char*)(e0 + (size_t)n * Fpad);
    const char* g1 = (const char*)(e1 + (size_t)n * Fpad);
    const char* g2 = (const char*)(e2 + (size_t)n * Fpad);
    const char* gz = (const char*)(ez + (size_t)n * Fpad);

    const unsigned lds0 = (unsigned)(size_t)(&smem[0]);

    // Cooperative async stage of one 16 KB chunk: 256 threads x 4 x 16B.
    auto stage = [&](int c, int buf) {
        unsigned lb = lds0 + (unsigned)(buf * CHUNK_BYTES) + (unsigned)(t * 16);
        size_t   go = (size_t)c * FUNC_BYTES + (size_t)(t * 16);
        asm volatile("global_load_async_to_lds_b128 %0, %1, off"
                     :: "v"(lb + 0u * FUNC_BYTES), "v"(g0 + go) : "memory");
        asm volatile("global_load_async_to_lds_b128 %0, %1, off"
                     :: "v"(lb + 1u * FUNC_BYTES), "v"(g1 + go) : "memory");
        asm volatile("global_load_async_to_lds_b128 %0, %1, off"
                     :: "v"(lb + 2u * FUNC_BYTES), "v"(g2 + go) : "memory");
        asm volatile("global_load_async_to_lds_b128 %0, %1, off"
                     :: "v"(lb + 3u * FUNC_BYTES), "v"(gz + go) : "memory");
    };

    // A-matrix: lanes 0-15 hold K0,K1 = (x, y); lanes 16-31 hold K2,K3 = (1, 0)
    v2f Apix;
    Apix.x = hi ? 1.0f : (float)(x0 + col);
    Apix.y = hi ? 0.0f : (float)row;

    float bestd[8];
    int   besti[8];
#pragma unroll
    for (int v = 0; v < 8; ++v) { bestd[v] = 1e30f; besti[v] = -1; }

    stage(0, 0);
    for (int c = 0; c < chunks; ++c) {
        const int buf = c & 1;
        if (c + 1 < chunks) {
            stage(c + 1, buf ^ 1);                       // overlap next DMA
            asm volatile("s_wait_asynccnt 0x4" ::: "memory");  // chunk c resident
        } else {
            asm volatile("s_wait_asynccnt 0x0" ::: "memory");
        }
        __syncthreads();                                 // publish chunk c

        const char* base = smem + buf * CHUNK_BYTES + col * 16 + hi * 8;
#pragma unroll 4
        for (int gi = 0; gi < GROUPS_PER_CHUNK; ++gi) {
            const char* p = base + gi * 256;
            v2f B0 = *(const v2f*)(p + 0 * FUNC_BYTES);
            v2f B1 = *(const v2f*)(p + 1 * FUNC_BYTES);
            v2f B2 = *(const v2f*)(p + 2 * FUNC_BYTES);
            v2f Bz = *(const v2f*)(p + 3 * FUNC_BYTES);
            v8f zacc = {};
            v8f d0 = __builtin_amdgcn_wmma_f32_16x16x4_f32(
                false, Apix, false, B0, (short)0, zacc, false, false);
            v8f d1 = __builtin_amdgcn_wmma_f32_16x16x4_f32(
                false, Apix, false, B1, (short)0, zacc, false, false);
            v8f d2 = __builtin_amdgcn_wmma_f32_16x16x4_f32(
                false, Apix, false, B2, (short)0, zacc, false, false);
            v8f dd = __builtin_amdgcn_wmma_f32_16x16x4_f32(
                false, Apix, false, Bz, (short)0, zacc, false, false);

            const int id0 = (((c << 4) + gi) << 4) + col;
#pragma unroll
            for (int v = 0; v < 8; ++v) {
                float m   = fminf(fminf(d0[v], d1[v]), d2[v]);   // v_min3
                float dep = dd[v];
                bool ok = (m >= 0.f) && (dep > 0.f) && (dep < bestd[v]);
                if (ok) { bestd[v] = dep; besti[v] = id0; }
            }
        }
        __syncthreads();     // readers done before buf is overwritten
    }

    // Min-reduce (depth, id) across the 16 triangle columns (xor masks 1..8
    // stay inside each 16-lane half). Tie-break: smaller face id.
#pragma unroll
    for (int m = 1; m < 16; m <<= 1) {
#pragma unroll
        for (int v = 0; v < 8; ++v) {
            float od = __shfl_xor(bestd[v], m, 32);
            int   oi = __shfl_xor(besti[v], m, 32);
            if (od < bestd[v] ||
                (od == bestd[v] && (unsigned)oi < (unsigned)besti[v])) {
                bestd[v] = od; besti[v] = oi;
            }
        }
    }

    // Lane half hi, slot v  <->  pixel M = v + 8*hi. Publish winners via LDS.
    if (lane == 0 || lane == 16) {
#pragma unroll
        for (int v = 0; v < 8; ++v) {
            sd[wave][hi * 8 + v] = bestd[v];
            si[wave][hi * 8 + v] = besti[v];
        }
    }
    __syncthreads();

    if (lane < 16) {
        const int x = x0 + lane;
        const int y = row;
        const int id = si[wave][lane];
        float o[RC];
#pragma unroll
        for (int c = 0; c < RC; ++c) o[c] = 0.f;

        if (id >= 0) {
            float4 f0 = e0[(size_t)n * Fpad + id];
            float4 f1 = e1[(size_t)n * Fpad + id];
            float fx = (float)x, fy = (float)y;
            float w0 = f0.x * fx + f0.y * fy + f0.z;
            float w1 = f1.x * fx + f1.y * fy + f1.z;
            float w2 = 1.0f - w0 - w1;
            int i0 = faces[id * 3 + 0];
            int i1 = faces[id * 3 + 1];
            int i2 = faces[id * 3 + 2];
            const float* cA = colors + ((size_t)n * V + i0) * RC;
            const float* cB = colors + ((size_t)n * V + i1) * RC;
            const float* cC = colors + ((size_t)n * V + i2) * RC;
            bool any = false;
#pragma unroll
            for (int c = 0; c < RC; ++c) {
                float val = w0 * cA[c] + w1 * cB[c] + w2 * cC[c];
                any = any || (val > 0.f);
                o[c] = val;
            }
            float mval = any ? 1.0f : 0.0f;
#pragma unroll
            for (int c = 0; c < RC; ++c) o[c] *= mval;
        }
#pragma unroll
        for (int c = 0; c < RC; ++c)
            out[(((size_t)n * RC + c) * RH + y) * RW + x] = o[c];
    }
}

// ---------------------------------------------------------------------------
extern "C" void kernel_launch(void* const* d_in, const int* in_sizes, int n_in,
                              void* d_out, int out_size, void* d_ws, size_t ws_size,
                              hipStream_t stream) {
    const float* verts  = (const float*)d_in[0];
    const float* colors = (const float*)d_in[1];
    const int*   faces  = (const int*)d_in[2];

    int F = in_sizes[2] / 3;                    // 13776
    int N = out_size / (RC * RH * RW);          // 2
    if (N < 1) N = 1;
    int V = in_sizes[0] / (N * 3);              // 6890
    int Fpad = (F + 255) & ~255;                // chunk = 256 triangles

    // Workspace: 4 coefficient tables of N*Fpad float4 (~1.8 MB total)
    float4* e0 = (float4*)d_ws;
    float4* e1 = e0 + (size_t)N * Fpad;
    float4* e2 = e1 + (size_t)N * Fpad;
    float4* ez = e2 + (size_t)N * Fpad;

    int tot = N * Fpad;
    raster_pre<<<(tot + 255) / 256, 256, 0, stream>>>(
        verts, faces, e0, e1, e2, ez, N, V, F, Fpad);

    dim3 grid(RW / 16, RH / 8, N);   // 16 x 32 x N blocks, 8 waves each
    raster_main<<<grid, 256, 0, stream>>>(
        colors, faces, e0, e1, e2, ez, (float*)d_out, V, Fpad);
}